// BeatsODE3_82265803587767
// MI455X (gfx1250) — compile-verified
//
#include <hip/hip_runtime.h>
#include <cstddef>

// ---------------------------------------------------------------------------
// BeatsODE3 for MI455X (gfx1250, wave32).
// Heavy op: repeated (512x512) @ (512 x M) f32 GEMMs (graph propagation) via
// V_WMMA_F32_16X16X4_F32. Propagation state is kept m-major (Z[m*512+v]) so
// the WMMA B fragment is a single contiguous b64 load per lane and the C/D
// epilogue is b128 traffic. Each wave: 1 v-tile x 4 m-subtiles with 4
// independent accumulators (hides WMMA->WMMA RAW hazard, reuses A 4x).
// adj (1MB) + state (<=5MB) live in the 192MB L2 -> compute-bound.
// ---------------------------------------------------------------------------

typedef float v2f __attribute__((ext_vector_type(2)));
typedef float v4f __attribute__((ext_vector_type(4)));
typedef float v8f __attribute__((ext_vector_type(8)));

#define CDIV(a, b) (((a) + (b) - 1) / (b))

static constexpr int B_     = 4;
static constexpr int N_     = 512;
static constexpr int CD     = 32;     // CONV_DIM
static constexpr int SEQ    = 12;
static constexpr int REC    = 26;     // RECEPTIVE
static constexpr int ENDD   = 128;    // END_DIM
static constexpr int MAXM   = 128 * 20;  // largest GEMM-N (L = 20)

// ---------------------------------------------------------------------------
// adjT[v*512+w] = adj[w*512+v]
// ---------------------------------------------------------------------------
__global__ void k_transpose512(const float* __restrict__ A, float* __restrict__ AT) {
  int idx = blockIdx.x * blockDim.x + threadIdx.x;
  if (idx >= N_ * N_) return;
  int w = idx % N_, v = idx / N_;
  AT[(size_t)v * N_ + w] = A[(size_t)w * N_ + v];
}

// ---------------------------------------------------------------------------
// start conv: z[b,o,n,l] = b[o] + sum_c W[o,c] * pad(x)[b,c,n,l],  pad = 14 left
// z layout (B,32,N,26) row-major
// ---------------------------------------------------------------------------
__global__ void k_start_conv(const float* __restrict__ x, const float* __restrict__ W,
                             const float* __restrict__ bias, float* __restrict__ z) {
  int idx = blockIdx.x * blockDim.x + threadIdx.x;
  const int total = B_ * CD * N_ * REC;
  if (idx >= total) return;
  int l = idx % REC;
  int n = (idx / REC) % N_;
  int o = (idx / (REC * N_)) % CD;
  int b = idx / (REC * N_ * CD);
  float v = bias[o];
  int xl = l - (REC - SEQ);
  if (xl >= 0) {
    v += W[o * 2 + 0] * x[((size_t)(b * 2 + 0) * N_ + n) * SEQ + xl]
       + W[o * 2 + 1] * x[((size_t)(b * 2 + 1) * N_ + n) * SEQ + xl];
  }
  z[idx] = v;
}

// ---------------------------------------------------------------------------
// inception gate h = tanh(inc1(xw)) * sigmoid(inc2(xw)), written DIRECTLY in
// m-major GEMM layout: zt0[((b*32+oc)*L + l)*512 + n], col m = (b*32+oc)*L+l.
// Thread index: n fastest -> coalesced stores. xw = z[..., -sl:].
// ---------------------------------------------------------------------------
struct IncParams {
  const float* w1[4]; const float* b1[4];
  const float* w2[4]; const float* b2[4];
};

__global__ void k_inception(const float* __restrict__ z, IncParams P,
                            float* __restrict__ zt0, int sl, int L) {
  int idx = blockIdx.x * blockDim.x + threadIdx.x;
  const int total = B_ * CD * N_ * L;
  if (idx >= total) return;
  int n  = idx % N_;
  int l  = (idx / N_) % L;
  int oc = (idx / (N_ * L)) % CD;
  int b  = idx / (N_ * L * CD);
  int br = oc >> 3, o = oc & 7;
  int k  = (br == 0) ? 2 : (br == 1) ? 3 : (br == 2) ? 6 : 7;

  const int zoff = (REC - sl) + (7 - k) + l;   // start pos in z last-dim
  const float* w1 = P.w1[br] + (size_t)o * CD * k;
  const float* w2 = P.w2[br] + (size_t)o * CD * k;
  float f = P.b1[br][o];
  float g = P.b2[br][o];
  for (int c = 0; c < CD; ++c) {
    const float* zp = z + ((size_t)(b * CD + c) * N_ + n) * REC + zoff;
    for (int u = 0; u < k; ++u) {
      float xv = zp[u];
      f += w1[c * k + u] * xv;
      g += w2[c * k + u] * xv;
    }
  }
  float hv = tanhf(f) * (1.f / (1.f + __expf(-g)));
  zt0[((size_t)(b * CD + oc) * L + l) * N_ + n] = hv;
}

// ---------------------------------------------------------------------------
// WMMA propagation step: Zout = 0.6*Zin + 0.4*(G @ Zin)
//   G: (512 x 512) row-major; Zin/Zout m-major: Z[m*512 + v], M % 64 == 0.
// One wave per (16 v) x (64 m) block = 4 WMMA tiles, 4 independent acc chains.
//   A 16x4 f32:  lane: row = lane%16, K = 2*(lane/16)+{0,1}  -> one b64 load
//   B 4x16 f32:  lane: col = lane%16, K = 2*(lane/16)+{0,1}  -> one b64 load
//   C/D 16x16:   VGPR r: row = v0 + r + 8*(lane/16), col = lane%16
//                -> rows contiguous in m-major => b128 epilogue
// ---------------------------------------------------------------------------
__global__ void __launch_bounds__(128) k_gemm_prop(const float* __restrict__ G,
                                                   const float* __restrict__ Zin,
                                                   float* __restrict__ Zout, int M) {
  const int wave = threadIdx.x >> 5;
  const int lane = threadIdx.x & 31;
  const int r16  = lane & 15;
  const int half = lane >> 4;
  const int mg   = M >> 6;                 // 64-wide m groups
  const int tile = blockIdx.x * 4 + wave;
  const int tv = tile / mg;
  const int tm = tile % mg;
  const int v0 = tv << 4;
  const int m0 = tm << 6;

  v8f acc[4] = {};
  const float* arow  = G   + (size_t)(v0 + r16) * N_ + 2 * half;
  const float* bbase = Zin + (size_t)(m0 + r16) * N_ + 2 * half;

  for (int k0 = 0; k0 < N_; k0 += 4) {
    v2f a = *(const v2f*)(arow + k0);
#pragma unroll
    for (int j = 0; j < 4; ++j) {
      v2f b = *(const v2f*)(bbase + (size_t)j * 16 * N_ + k0);
      acc[j] = __builtin_amdgcn_wmma_f32_16x16x4_f32(
          /*neg_a=*/false, a, /*neg_b=*/false, b,
          /*c_mod=*/(short)0, acc[j], /*reuse_a=*/false, /*reuse_b=*/false);
    }
  }

#pragma unroll
  for (int j = 0; j < 4; ++j) {
    size_t base = ((size_t)(m0 + j * 16 + r16)) * N_ + v0 + 8 * half;
    v4f zi0 = *(const v4f*)(Zin + base);
    v4f zi1 = *(const v4f*)(Zin + base + 4);
    v4f lo, hi;
#pragma unroll
    for (int r = 0; r < 4; ++r) {
      lo[r] = 0.6f * zi0[r] + 0.4f * acc[j][r];
      hi[r] = 0.6f * zi1[r] + 0.4f * acc[j][r + 4];
    }
    *(v4f*)(Zout + base)     = lo;
    *(v4f*)(Zout + base + 4) = hi;
  }
}

// ---------------------------------------------------------------------------
// combine: f = conv1x1(cat(zt0..3), g1) + conv1x1(cat(wt0..3), g2);
//          z[..., 26-L+l] += H1 * f      (H1 = 0.25)
// zt/wt are m-major: zt[((b*32+c)*L + l)*512 + n]. Thread n fastest ->
// coalesced on the 256 reads/thread; single strided write to z.
// ---------------------------------------------------------------------------
struct GcParams { const float* zt[4]; const float* wt[4]; };

__global__ void k_gcombine(GcParams P,
                           const float* __restrict__ g1w, const float* __restrict__ g1b,
                           const float* __restrict__ g2w, const float* __restrict__ g2b,
                           float* __restrict__ z, int L) {
  int idx = blockIdx.x * blockDim.x + threadIdx.x;
  const int total = B_ * CD * N_ * L;
  if (idx >= total) return;
  int n = idx % N_;
  int l = (idx / N_) % L;
  int o = (idx / (N_ * L)) % CD;
  int b = idx / (N_ * L * CD);
  size_t colbase = ((size_t)(b * CD) * L + l) * N_ + n;   // + c*L*512 per channel

  float s = g1b[o] + g2b[o];
  for (int i = 0; i < 4; ++i) {
    const float* Z1 = P.zt[i];
    const float* Z2 = P.wt[i];
    const float* W1 = g1w + (size_t)o * 128 + i * CD;
    const float* W2 = g2w + (size_t)o * 128 + i * CD;
    for (int c = 0; c < CD; ++c) {
      size_t ci = colbase + (size_t)c * L * N_;
      s += W1[c] * Z1[ci] + W2[c] * Z2[ci];
    }
  }
  z[((size_t)(b * CD + o) * N_ + n) * REC + (REC - L) + l] += 0.25f * s;
}

// ---------------------------------------------------------------------------
// layernorm over (c,n,l) of z[..., -2:] per batch -> zn (B,32,N,2)
// ---------------------------------------------------------------------------
__global__ void k_layernorm(const float* __restrict__ z, float* __restrict__ zn) {
  const int b = blockIdx.x;
  const int t = threadIdx.x;
  __shared__ float ssum[256], ssq[256];
  const int NE = CD * N_ * 2;                 // 32768
  float s = 0.f, q = 0.f;
  for (int i = t; i < NE; i += 256) {
    int l = i & 1, n = (i >> 1) & (N_ - 1), c = i >> 10;
    float v = z[((size_t)(b * CD + c) * N_ + n) * REC + (REC - 2) + l];
    s += v; q += v * v;
  }
  ssum[t] = s; ssq[t] = q;
  __syncthreads();
  for (int st = 128; st > 0; st >>= 1) {
    if (t < st) { ssum[t] += ssum[t + st]; ssq[t] += ssq[t + st]; }
    __syncthreads();
  }
  float mean = ssum[0] / (float)NE;
  float var  = ssq[0] / (float)NE - mean * mean;
  float inv  = rsqrtf(var + 1e-5f);
  for (int i = t; i < NE; i += 256) {
    int l = i & 1, n = (i >> 1) & (N_ - 1), c = i >> 10;
    float v = z[((size_t)(b * CD + c) * N_ + n) * REC + (REC - 2) + l];
    zn[((size_t)(b * CD + c) * N_ + n) * 2 + l] = (v - mean) * inv;
  }
}

// ---------------------------------------------------------------------------
// head stage 1: e1[b,o,n,l] = relu(b1[o] + sum_c W[o,c]*zn[b,c,n,l]), o<128,l<2
// ---------------------------------------------------------------------------
__global__ void k_head1(const float* __restrict__ zn, const float* __restrict__ W,
                        const float* __restrict__ bias, float* __restrict__ e1) {
  int idx = blockIdx.x * blockDim.x + threadIdx.x;
  const int total = B_ * ENDD * N_ * 2;
  if (idx >= total) return;
  int l = idx & 1;
  int n = (idx >> 1) % N_;
  int o = (idx / (2 * N_)) % ENDD;
  int b = idx / (2 * N_ * ENDD);
  float s = bias[o];
  const float* wr = W + (size_t)o * CD;
  const float* zp = zn + ((size_t)b * CD * N_ + n) * 2 + l;   // c-stride = N_*2
  for (int c = 0; c < CD; ++c) s += wr[c] * zp[(size_t)c * N_ * 2];
  e1[idx] = fmaxf(s, 0.f);
}

// ---------------------------------------------------------------------------
// head stage 2 (+ output transpose): acc[b,c,n,t] += scale*(b2[t] +
//    sum_o W2[t,o] * e1[b,o,n,l=c]),  acc layout (B,2,N,T)
// ---------------------------------------------------------------------------
__global__ void k_head2_add(const float* __restrict__ e1, const float* __restrict__ W,
                            const float* __restrict__ bias, float* __restrict__ acc,
                            int T, float scale) {
  int idx = blockIdx.x * blockDim.x + threadIdx.x;
  const int total = B_ * 2 * N_ * T;
  if (idx >= total) return;
  int t = idx % T;
  int n = (idx / T) % N_;
  int c = (idx / (T * N_)) % 2;
  int b = idx / (T * N_ * 2);
  float s = bias[t];
  const float* wr = W + (size_t)t * ENDD;
  const float* ep = e1 + ((size_t)b * ENDD * N_ + n) * 2 + c;  // o-stride = N_*2
  for (int o = 0; o < ENDD; ++o) s += wr[o] * ep[(size_t)o * N_ * 2];
  acc[idx] += scale * s;
}

// ---------------------------------------------------------------------------
// final output: out[b,si,n,c] = fc_acc[b,c,n,si]
// ---------------------------------------------------------------------------
__global__ void k_write_out(const float* __restrict__ acc, float* __restrict__ out, int S) {
  int idx = blockIdx.x * blockDim.x + threadIdx.x;
  const int total = B_ * S * N_ * 2;
  if (idx >= total) return;
  int c  = idx & 1;
  int n  = (idx >> 1) % N_;
  int si = (idx / (2 * N_)) % S;
  int b  = idx / (2 * N_ * S);
  out[idx] = acc[((size_t)(b * 2 + c) * N_ + n) * S + si];
}

// ---------------------------------------------------------------------------
// host orchestration
// ---------------------------------------------------------------------------
extern "C" void kernel_launch(void* const* d_in, const int* in_sizes, int n_in,
                              void* d_out, int out_size, void* d_ws, size_t ws_size,
                              hipStream_t stream) {
  const float* backcast = (const float*)d_in[0];
  const float* adj      = (const float*)d_in[1];
  // params: 3 stacks x 30 leaves, reference-source insertion order:
  //  0 start_w  1 start_b  2..5 inc1_w  6..9 inc1_b  10..13 inc2_w  14..17 inc2_b
  //  18 g1_w 19 g1_b 20 g2_w 21 g2_b  22 bc1_w 23 bc1_b 24 bc2_w 25 bc2_b
  //  26 fc1_w 27 fc1_b 28 fc2_w 29 fc2_b
  auto prm = [&](int s, int j) -> const float* { return (const float*)d_in[2 + s * 30 + j]; };

  // workspace layout (floats)
  size_t off = 0;
  auto alloc = [&](size_t n) { float* p = (float*)d_ws + off; off += n; return p; };
  float* adjT = alloc((size_t)N_ * N_);
  float* z    = alloc((size_t)B_ * CD * N_ * REC);
  float* x    = alloc((size_t)B_ * 2 * N_ * SEQ);
  float* facc = alloc((size_t)B_ * 2 * N_ * SEQ);
  float* zn   = alloc((size_t)B_ * CD * N_ * 2);
  float* e1   = alloc((size_t)B_ * ENDD * N_ * 2);
  float* zt[4];
  float* wt[4];
  zt[0] = alloc((size_t)N_ * MAXM);
  for (int i = 1; i < 4; ++i) zt[i] = alloc((size_t)N_ * MAXM);
  wt[0] = zt[0];
  for (int i = 1; i < 4; ++i) wt[i] = alloc((size_t)N_ * MAXM);
  (void)ws_size; (void)in_sizes; (void)n_in;

  const int T = 256;
  const int SLS[3] = {3, 6, 12};

  k_transpose512<<<CDIV(N_ * N_, T), T, 0, stream>>>(adj, adjT);
  hipMemcpyAsync(x, backcast, (size_t)B_ * 2 * N_ * SEQ * sizeof(float),
                 hipMemcpyDeviceToDevice, stream);

  size_t out_off = 0;
  for (int s = 0; s < 3; ++s) {
    const int S = SLS[s];
    hipMemsetAsync(facc, 0, (size_t)B_ * 2 * N_ * S * sizeof(float), stream);

    IncParams IP;
    for (int i = 0; i < 4; ++i) {
      IP.w1[i] = prm(s, 2 + i);  IP.b1[i] = prm(s, 6 + i);
      IP.w2[i] = prm(s, 10 + i); IP.b2[i] = prm(s, 14 + i);
    }
    GcParams GP;
    for (int i = 0; i < 4; ++i) { GP.zt[i] = zt[i]; GP.wt[i] = wt[i]; }

    for (int it = 0; it < 3; ++it) {                 // NFE0
      // --- beats_eval ---
      k_start_conv<<<CDIV(B_ * CD * N_ * REC, T), T, 0, stream>>>(
          x, prm(s, 0), prm(s, 1), z);

      int sl = REC;
      for (int blk = 0; blk < 4; ++blk) {            // NFE1 st_blocks
        const int L = sl - 6;
        const int M = 128 * L;                       // 2560,1792,1024,256: all %64==0
        k_inception<<<CDIV(B_ * CD * N_ * L, T), T, 0, stream>>>(z, IP, zt[0], sl, L);

        const int tiles = (N_ / 16) * (M / 64);      // 16v x 64m waves
        const int gb = tiles / 4;                    // 4 waves/block, exact
        // branch 1: G = adj
        k_gemm_prop<<<gb, 128, 0, stream>>>(adj,  zt[0], zt[1], M);
        k_gemm_prop<<<gb, 128, 0, stream>>>(adj,  zt[1], zt[2], M);
        k_gemm_prop<<<gb, 128, 0, stream>>>(adj,  zt[2], zt[3], M);
        // branch 2: G = adj^T
        k_gemm_prop<<<gb, 128, 0, stream>>>(adjT, wt[0], wt[1], M);
        k_gemm_prop<<<gb, 128, 0, stream>>>(adjT, wt[1], wt[2], M);
        k_gemm_prop<<<gb, 128, 0, stream>>>(adjT, wt[2], wt[3], M);

        k_gcombine<<<CDIV(B_ * CD * N_ * L, T), T, 0, stream>>>(
            GP, prm(s, 18), prm(s, 19), prm(s, 20), prm(s, 21), z, L);
        sl -= 6;
      }

      k_layernorm<<<B_, 256, 0, stream>>>(z, zn);

      // backcast head: x += H0 * bc   (H0 = 0.4)
      k_head1<<<CDIV(B_ * ENDD * N_ * 2, T), T, 0, stream>>>(zn, prm(s, 22), prm(s, 23), e1);
      k_head2_add<<<CDIV(B_ * 2 * N_ * SEQ, T), T, 0, stream>>>(
          e1, prm(s, 24), prm(s, 25), x, SEQ, 0.4f);
      // forecast head: facc += fc
      k_head1<<<CDIV(B_ * ENDD * N_ * 2, T), T, 0, stream>>>(zn, prm(s, 26), prm(s, 27), e1);
      k_head2_add<<<CDIV(B_ * 2 * N_ * S, T), T, 0, stream>>>(
          e1, prm(s, 28), prm(s, 29), facc, S, 1.0f);
    }

    k_write_out<<<CDIV(B_ * S * N_ * 2, T), T, 0, stream>>>(
        facc, (float*)d_out + out_off, S);
    out_off += (size_t)B_ * S * N_ * 2;
  }
  (void)out_size;
}